// MultiModalModel_31774168055837
// MI455X (gfx1250) — compile-verified
//
#include <hip/hip_runtime.h>

typedef __attribute__((ext_vector_type(2))) float v2f;
typedef __attribute__((ext_vector_type(8))) float v8f;

#define I_N 256
#define T_N 256
#define L_N 32
#define E_N 128
#define P_N 49     // H*W patches
#define PPAD 64    // padded patch count (4 WMMA N-tiles)
#define KP_N 64    // E/2 interleaved k-row pairs
#define PSTR 80    // float2 stride per kpair row: 160 dwords == 32 mod 64 -> halves hit disjoint banks

// One block = 1 image x 128 (t,l) rows (= 4 texts). 8 waves, each wave owns a
// 16-row x 64-col sim tile accumulated in fp32 WMMA over K = E = 128.
// LDS image tile stored fragment-native: lB[kp][p] = (img[2kp][p], img[2kp+1][p])
// so each B fragment is one ds_load_b64 into an even-aligned VGPR pair.
__global__ __launch_bounds__(256)
void clip_patch_logits_kernel(const float* __restrict__ img,
                              const float* __restrict__ txt,
                              const int*   __restrict__ tlen,
                              const float* __restrict__ nlt,
                              float* __restrict__ out)
{
    __shared__ v2f   lB[KP_N * PSTR];   // 40960 bytes
    __shared__ float partials[8];

    const int bx    = blockIdx.x;
    const int image = bx >> 6;          // 256 images
    const int tb    = bx & 63;          // 64 text blocks of 128 (t,l) rows
    const int tid   = threadIdx.x;

    // --- stage image block as k-pair-interleaved [64][64] float2 (zeros p>=49) ---
    const float* ibase = img + (size_t)image * (E_N * P_N);
    for (int idx = tid; idx < KP_N * PPAD; idx += 256) {
        const int kp = idx >> 6;
        const int p  = idx & 63;
        v2f v = {};
        if (p < P_N) {
            v.x = ibase[(2 * kp)     * P_N + p];
            v.y = ibase[(2 * kp + 1) * P_N + p];
        }
        lB[kp * PSTR + p] = v;
    }
    __syncthreads();

    const int wave = tid >> 5;
    const int lane = tid & 31;
    const int hi   = lane >> 4;         // which half-wave (selects K pair)
    const int mn   = lane & 15;         // A: row m, B: col n

    // A-fragment source: text row (t*L + l), contiguous in e
    const int row = tb * 128 + wave * 16 + mn;
    const float* arow = txt + (size_t)row * E_N;

    v8f acc0 = {}, acc1 = {}, acc2 = {}, acc3 = {};

    #pragma unroll 4
    for (int kk = 0; kk < E_N / 4; ++kk) {
        const int k = kk * 4 + hi * 2;
        v2f a;                          // A 16x4: lane m=lane%16, k = 2*(lane/16)+v
        a.x = arow[k];
        a.y = arow[k + 1];
        // B 4x16: lane n=lane%16, rows (k, k+1) come as one float2
        const v2f* brow = &lB[(2 * kk + hi) * PSTR + mn];
        acc0 = __builtin_amdgcn_wmma_f32_16x16x4_f32(false, a, false, brow[0],  (short)0, acc0, false, false);
        acc1 = __builtin_amdgcn_wmma_f32_16x16x4_f32(false, a, false, brow[16], (short)0, acc1, false, false);
        acc2 = __builtin_amdgcn_wmma_f32_16x16x4_f32(false, a, false, brow[32], (short)0, acc2, false, false);
        acc3 = __builtin_amdgcn_wmma_f32_16x16x4_f32(false, a, false, brow[48], (short)0, acc3, false, false);
    }

    // D layout: VGPR r -> lanes 0-15: (M=r, N=lane); lanes 16-31: (M=r+8, N=lane-16)
    // max over patches (tile 3: only N==0, i.e. p=48, is real data), then sum rows.
    float rowsum = 0.0f;
    #pragma unroll
    for (int r = 0; r < 8; ++r) {
        float v = fmaxf(fmaxf(acc0[r], acc1[r]), acc2[r]);
        v = fmaxf(v, (mn == 0) ? acc3[r] : -3.402823466e38f);
        #pragma unroll
        for (int off = 1; off <= 8; off <<= 1)
            v = fmaxf(v, __shfl_xor(v, off, 32));   // max within each 16-lane half
        rowsum += v;   // lanes 0-15 accumulate rows 0..7, lanes 16-31 rows 8..15
    }
    rowsum += __shfl_xor(rowsum, 16, 32);           // full 16-row sum, all lanes
    if (lane == 0) partials[wave] = rowsum;
    __syncthreads();

    // waves (2j, 2j+1) cover the 32 tokens of text tb*4+j
    if (tid < 4) {
        const float total = partials[2 * tid] + partials[2 * tid + 1];
        const int   t     = tb * 4 + tid;
        const float scale = expf(nlt[0]);
        const float val   = total / (float)tlen[t] * scale;
        out[(size_t)image * T_N + t] = val;                       // logits_per_image [I,T]
        out[(size_t)I_N * T_N + (size_t)t * I_N + image] = val;   // logits_per_text  [T,I]
    }
}

extern "C" void kernel_launch(void* const* d_in, const int* in_sizes, int n_in,
                              void* d_out, int out_size, void* d_ws, size_t ws_size,
                              hipStream_t stream) {
    (void)in_sizes; (void)n_in; (void)d_ws; (void)ws_size; (void)out_size;
    const float* image_features = (const float*)d_in[0];  // [256,128,7,7] f32
    const float* text_features  = (const float*)d_in[1];  // [256,32,128]  f32
    const int*   text_length    = (const int*)d_in[2];    // [256]         i32
    const float* nlt            = (const float*)d_in[3];  // scalar        f32
    float* out = (float*)d_out;                           // [256*256]*2   f32

    dim3 grid(I_N * 64);   // 256 images x 64 text blocks
    dim3 block(256);
    clip_patch_logits_kernel<<<grid, block, 0, stream>>>(
        image_features, text_features, text_length, nlt, out);
}